// LIFLayer_41248865910902
// MI455X (gfx1250) — compile-verified
//
#include <hip/hip_runtime.h>
#include <hip/hip_bf16.h>

typedef __attribute__((ext_vector_type(2))) float v2f;
typedef __attribute__((ext_vector_type(8))) float v8f;

#define N_DIM 4096
#define I_DIM 4096
#define B_DIM 256

__device__ __constant__ const float c_DT     = 1.0f;
__device__ __constant__ const float c_THRESH = 0.6f;
__device__ __constant__ const float c_REFRAC = 0.0f;
__device__ __constant__ const float c_GAMMA  = 0.3f;
__device__ __constant__ const float c_ALPHA  = 0.1f;   // DT/TAU

static __device__ __forceinline__ v8f wmma_f32(v2f a, v2f b, v8f c) {
    // 8 args: (neg_a, A, neg_b, B, c_mod, C, reuse_a, reuse_b)
    return __builtin_amdgcn_wmma_f32_16x16x4_f32(false, a, false, b, (short)0, c,
                                                 false, false);
}

// ---------------------------------------------------------------------------
// Kernel 1: new_v = W @ inputs, fused with LIF state update.
// Block = 256 threads = 8 waves. Block owns M-tile of 16 rows x ALL 256 batch
// columns (wave w covers columns [w*32, w*32+32) via two 16x16 accumulators),
// so W rows are streamed from HBM exactly once across the whole dispatch.
// A layout (16x4 f32): lanes 0-15 = rows M, VGPR0/1 = K,K+1; half-wave +2 in K.
// B layout (4x16 f32): VGPR v, half h -> K = k0 + v + 2h, lane%16 -> column.
// ---------------------------------------------------------------------------
__global__ __launch_bounds__(256) void lif_gemm_lif_kernel(
    const float* __restrict__ W, const float* __restrict__ inp,
    const float* __restrict__ u, const float* __restrict__ r,
    float* __restrict__ out_z, float* __restrict__ out_u,
    float* __restrict__ out_r, float* __restrict__ out_h)
{
    const int lane = threadIdx.x & 31;
    const int wave = threadIdx.x >> 5;
    const int half = lane >> 4;      // 0 or 1 (half-wave)
    const int l16  = lane & 15;
    const int m0   = blockIdx.x * 16;
    const int n0   = wave * 32;      // two 16-wide column tiles: n0, n0+16

    const float* Wrow = W + (size_t)(m0 + l16) * I_DIM;
    v8f c0 = {};
    v8f c1 = {};

    for (int k0 = 0; k0 < I_DIM; k0 += 16) {
        if (k0 + 1024 < I_DIM)
            __builtin_prefetch(Wrow + k0 + 1024, 0, 0);  // global_prefetch_b8
#pragma unroll
        for (int kk = 0; kk < 16; kk += 4) {
            const int ko = k0 + kk + 2 * half;
            v2f a = *(const v2f*)(Wrow + ko);               // K=ko, ko+1
            const float* bp = inp + (size_t)ko * B_DIM;
            v2f b0, b1;
            b0.x = bp[n0 + l16];                            // K=ko
            b0.y = bp[B_DIM + n0 + l16];                    // K=ko+1
            b1.x = bp[n0 + 16 + l16];
            b1.y = bp[B_DIM + n0 + 16 + l16];
            c0 = wmma_f32(a, b0, c0);
            c1 = wmma_f32(a, b1, c1);
        }
    }

    // Fused LIF epilogue directly from accumulator layout:
    // VGPR rr, half h -> row = m0 + rr + 8h; lane%16 -> column.
#pragma unroll
    for (int t = 0; t < 2; ++t) {
        v8f c = t ? c1 : c0;
        const int col = n0 + t * 16 + l16;
#pragma unroll
        for (int rr = 0; rr < 8; ++rr) {
            const int row = m0 + rr + 8 * half;
            const size_t idx = (size_t)row * B_DIM + col;
            const float nv = c[rr];
            const float uu = u[idx];
            const float ro = r[idx];
            const bool resting = (ro - c_DT) > 0.0f;
            float nu = resting ? uu : fmaf(nv, c_ALPHA, uu);
            float nr = resting ? (ro - c_DT) : 0.0f;
            const bool firing = nu > c_THRESH;
            const float z = firing ? 1.0f : 0.0f;
            nu = firing ? (nu - c_THRESH) : nu;
            nr = firing ? c_REFRAC : nr;
            const float h = c_GAMMA *
                fmaxf(0.0f, 1.0f - fabsf((nu - c_THRESH) * (1.0f / c_THRESH)));
            out_z[idx] = z;
            out_u[idx] = nu;
            out_r[idx] = nr;
            out_h[idx] = h;
        }
    }
}

// ---------------------------------------------------------------------------
// Kernel 2: new_eps = ALPHA * eps + inputs   ([I, B] elementwise)
// ---------------------------------------------------------------------------
__global__ __launch_bounds__(256) void eps_update_kernel(
    const float* __restrict__ inp, const float* __restrict__ eps,
    float* __restrict__ out_eps)
{
    const size_t idx = (size_t)blockIdx.x * blockDim.x + threadIdx.x;
    if (idx < (size_t)I_DIM * B_DIM)
        out_eps[idx] = fmaf(c_ALPHA, eps[idx], inp[idx]);
}

// ---------------------------------------------------------------------------
// Kernel 3: new_e[n,i] = sum_b h[n,b] * new_eps[i,b]  (K = 256)
// Both operands are row-major over the contraction dim b, so A and B tiles
// are single contiguous v2f loads per lane. Block = 8 waves, 16 rows x 256
// cols of the [4096 x 4096] output; grid = (256, 16).
// ---------------------------------------------------------------------------
__global__ __launch_bounds__(256) void etrace_gemm_kernel(
    const float* __restrict__ h, const float* __restrict__ eps2,
    float* __restrict__ e)
{
    const int lane = threadIdx.x & 31;
    const int wave = threadIdx.x >> 5;
    const int half = lane >> 4;
    const int l16  = lane & 15;
    const int m0   = blockIdx.x * 16;
    const int j0   = blockIdx.y * 256 + wave * 32;

    const float* Arow = h    + (size_t)(m0 + l16) * B_DIM;
    const float* Bc0  = eps2 + (size_t)(j0 + l16) * B_DIM;
    const float* Bc1  = eps2 + (size_t)(j0 + 16 + l16) * B_DIM;

    v8f c0 = {};
    v8f c1 = {};
#pragma unroll 4
    for (int k0 = 0; k0 < B_DIM; k0 += 4) {
        const int ko = k0 + 2 * half;
        v2f a  = *(const v2f*)(Arow + ko);
        v2f b0 = *(const v2f*)(Bc0 + ko);
        v2f b1 = *(const v2f*)(Bc1 + ko);
        c0 = wmma_f32(a, b0, c0);
        c1 = wmma_f32(a, b1, c1);
    }

#pragma unroll
    for (int t = 0; t < 2; ++t) {
        v8f c = t ? c1 : c0;
        const int col = j0 + t * 16 + l16;
#pragma unroll
        for (int rr = 0; rr < 8; ++rr) {
            const int row = m0 + rr + 8 * half;
            e[(size_t)row * I_DIM + col] = c[rr];
        }
    }
}

// ---------------------------------------------------------------------------
// Host launcher.
// Inputs (setup_inputs order): inputs[I,B], W[N,I], u[N,B], r[N,B],
//                              e[N,I] (unused), eps[I,B]
// Outputs (return order, concat): z[N,B], new_u[N,B], new_r[N,B],
//                                 new_e[N,I], new_eps[I,B], h[N,B]
// ---------------------------------------------------------------------------
extern "C" void kernel_launch(void* const* d_in, const int* in_sizes, int n_in,
                              void* d_out, int out_size, void* d_ws, size_t ws_size,
                              hipStream_t stream) {
    (void)in_sizes; (void)n_in; (void)out_size; (void)d_ws; (void)ws_size;

    const float* inp = (const float*)d_in[0];
    const float* W   = (const float*)d_in[1];
    const float* u   = (const float*)d_in[2];
    const float* r   = (const float*)d_in[3];
    const float* eps = (const float*)d_in[5];

    const size_t NB = (size_t)N_DIM * B_DIM;
    const size_t NI = (size_t)N_DIM * I_DIM;
    const size_t IB = (size_t)I_DIM * B_DIM;

    float* out   = (float*)d_out;
    float* o_z   = out;
    float* o_u   = out + NB;
    float* o_r   = out + 2 * NB;
    float* o_e   = out + 3 * NB;
    float* o_eps = out + 3 * NB + NI;
    float* o_h   = out + 3 * NB + NI + IB;

    // GEMM1 + LIF: grid = N/16 M-tiles, each block covers all 256 batch cols.
    lif_gemm_lif_kernel<<<dim3(N_DIM / 16, 1, 1), dim3(256, 1, 1), 0, stream>>>(
        W, inp, u, r, o_z, o_u, o_r, o_h);

    // eps decay (independent of kernel 1).
    eps_update_kernel<<<dim3((unsigned)(IB / 256), 1, 1), dim3(256, 1, 1), 0, stream>>>(
        inp, eps, o_eps);

    // Eligibility-trace GEMM: consumes h and new_eps produced above.
    etrace_gemm_kernel<<<dim3(N_DIM / 16, I_DIM / 256, 1), dim3(256, 1, 1), 0, stream>>>(
        o_h, o_eps, o_e);
}